// GnnAgent_29214367547977
// MI455X (gfx1250) — compile-verified
//
#include <hip/hip_runtime.h>
#include <hip/hip_bf16.h>

// GNN agent step on MI455X (gfx1250, wave32).
// f32 WMMA 16x16x4 keeps reference numerics. Edge GEMM restructured into
// per-node message precompute + L2-resident gather/scatter-add.
// Round-2 optimizations: weights pre-packed into WMMA B-fragment order
// (1 coalesced b64 load per fragment instead of 2 strided b32), and 2-way
// row blocking so each B fragment feeds 2 WMMAs (~0.8 loads/WMMA in GRU).

typedef __attribute__((ext_vector_type(2))) float v2f;
typedef __attribute__((ext_vector_type(8))) float v8f;

// ---- WMMA f32 16x16x4 --------------------------------------------------
__device__ __forceinline__ v8f wmma4(v2f a, v2f b, v8f c) {
    return __builtin_amdgcn_wmma_f32_16x16x4_f32(
        false, a, false, b, (short)0, c, false, false);
}

// A-fragment (16x4): lane L: row = row0+(L&15), k = k0+2*(L>>4) -> b64 load
__device__ __forceinline__ v2f load_a_frag(const float* M, int ld, int row0,
                                           int k0, int lane) {
    int r = row0 + (lane & 15);
    int k = k0 + ((lane >> 4) << 1);
    const float* p = M + (size_t)r * ld + k;
    v2f a; a.x = p[0]; a.y = p[1];
    return a;
}

// Packed B-fragment: weight pre-arranged so each (kt, nt) 4x16 tile is a
// contiguous 64-float block in lane order -> one coalesced b64 per lane.
__device__ __forceinline__ v2f load_b_packed(const float* P, int nk, int kt,
                                             int nt, int lane) {
    const v2f* p = (const v2f*)P + ((size_t)(nt * nk + kt) * 32 + lane);
    return *p;
}

__device__ __forceinline__ float sigmoid_f(float v) {
    return 1.0f / (1.0f + __expf(-v));
}
__device__ __forceinline__ float tanh_f(float v) {
    float t = __expf(-2.0f * v);
    return (1.0f - t) / (1.0f + t);
}

// ---- pack: W[K x Nc] row-major -> fragment-ordered blocks ---------------
// One wave per (kt, nt) tile; waves = (K/4)*(Nc/16).
__global__ __launch_bounds__(256) void pack_kernel(
    const float* __restrict__ W, float* __restrict__ P, int K, int Nc) {
    int w = (int)((blockIdx.x * 256u + threadIdx.x) >> 5);
    int lane = threadIdx.x & 31;
    int nk = K >> 2, nn = Nc >> 4;
    if (w >= nk * nn) return;
    int kt = w % nk, nt = w / nk;
    int n = nt * 16 + (lane & 15);
    int k = kt * 4 + ((lane >> 4) << 1);
    v2f v;
    v.x = W[(size_t)k * Nc + n];
    v.y = W[(size_t)(k + 1) * Nc + n];
    ((v2f*)P)[(size_t)(nt * nk + kt) * 32 + lane] = v;
}

// ---- zero scratch ------------------------------------------------------
__global__ __launch_bounds__(256) void zero_kernel(float* p, long n) {
    long i = (long)blockIdx.x * blockDim.x + threadIdx.x;
    long stride = (long)gridDim.x * blockDim.x;
    for (; i < n; i += stride) p[i] = 0.0f;
}

// ---- encoder  x = relu(relu(feat@W0+b0)@W1+b1) -------------------------
// 4 waves/block, 16 rows/wave; stage-1 result staged in 64 KB LDS.
__global__ __launch_bounds__(128) void enc_kernel(
    const float* __restrict__ feat, const float* __restrict__ pW0,
    const float* __restrict__ b0, const float* __restrict__ pW1,
    const float* __restrict__ b1, float* __restrict__ x, int N) {
    __shared__ float t[4][16 * 256];   // 64 KB
    const int wave = threadIdx.x >> 5;
    const int lane = threadIdx.x & 31;
    const int row0 = (blockIdx.x * 4 + wave) * 16;
    const bool active = (row0 < N);

    if (active) {
        v2f afr[16];
#pragma unroll
        for (int kk = 0; kk < 16; ++kk)
            afr[kk] = load_a_frag(feat, 64, row0, kk * 4, lane);
#pragma unroll 1
        for (int nt = 0; nt < 16; ++nt) {
            v8f acc = {};
#pragma unroll
            for (int kk = 0; kk < 16; ++kk) {
                v2f b = load_b_packed(pW0, 16, kk, nt, lane);
                acc = wmma4(afr[kk], b, acc);
            }
            int col = nt * 16 + (lane & 15);
            float bias = b0[col];
#pragma unroll
            for (int i = 0; i < 8; ++i) {
                int r = i + 8 * (lane >> 4);
                t[wave][r * 256 + col] = fmaxf(acc[i] + bias, 0.0f);
            }
        }
    }
    __syncthreads();
    if (active) {
#pragma unroll 1
        for (int nt = 0; nt < 16; ++nt) {
            v8f acc = {};
#pragma unroll 4
            for (int kk = 0; kk < 64; ++kk) {
                int k = kk * 4 + ((lane >> 4) << 1);
                int r = lane & 15;
                v2f a; a.x = t[wave][r * 256 + k]; a.y = t[wave][r * 256 + k + 1];
                v2f b = load_b_packed(pW1, 64, kk, nt, lane);
                acc = wmma4(a, b, acc);
            }
            int col = nt * 16 + (lane & 15);
            float bias = b1[col];
#pragma unroll
            for (int i = 0; i < 8; ++i) {
                int r = row0 + i + 8 * (lane >> 4);
                x[(size_t)r * 256 + col] = fmaxf(acc[i] + bias, 0.0f);
            }
        }
    }
}

// ---- per-node message  mx = [x|h] @ msg_W + msg_b  (K=512, R=2) --------
__global__ __launch_bounds__(128) void msg_kernel(
    const float* __restrict__ x, const float* __restrict__ h,
    const float* __restrict__ pmsgW, const float* __restrict__ msgb,
    float* __restrict__ mx, int N) {
    const int wave = threadIdx.x >> 5;
    const int lane = threadIdx.x & 31;
    const int row0 = (blockIdx.x * 4 + wave) * 32;   // two 16-row tiles
    if (row0 >= N) return;
#pragma unroll 1
    for (int nt = 0; nt < 8; ++nt) {
        v8f acc0 = {}, acc1 = {};
#pragma unroll 4
        for (int kk = 0; kk < 128; ++kk) {           // K = 512
            int k0 = kk * 4;
            v2f b = load_b_packed(pmsgW, 128, kk, nt, lane);
            v2f a0 = (k0 < 256) ? load_a_frag(x, 256, row0, k0, lane)
                                : load_a_frag(h, 256, row0, k0 - 256, lane);
            v2f a1 = (k0 < 256) ? load_a_frag(x, 256, row0 + 16, k0, lane)
                                : load_a_frag(h, 256, row0 + 16, k0 - 256, lane);
            acc0 = wmma4(a0, b, acc0);
            acc1 = wmma4(a1, b, acc1);
        }
        int col = nt * 16 + (lane & 15);
        float bias = msgb[col];
#pragma unroll
        for (int i = 0; i < 8; ++i) {
            int r = row0 + i + 8 * (lane >> 4);
            mx[(size_t)r * 128 + col] = acc0[i] + bias;
            mx[(size_t)(r + 16) * 128 + col] = acc1[i] + bias;
        }
    }
}

// ---- edge scatter: sums[dst] += mx[src]; cnt[dst] += 1 -----------------
__global__ __launch_bounds__(256) void edge_kernel(
    const int* __restrict__ src, const int* __restrict__ dst,
    const float* __restrict__ mx, float* __restrict__ sums,
    float* __restrict__ cnt, int E) {
    int e = (int)((blockIdx.x * 256u + threadIdx.x) >> 5);
    int lane = threadIdx.x & 31;
    if (e >= E) return;
    int s = src[e];
    int d = dst[e];
    const float4* mv = (const float4*)(mx + (size_t)s * 128);
    float4 v = mv[lane];
    float* base = sums + (size_t)d * 128 + lane * 4;
    atomicAdd(base + 0, v.x);
    atomicAdd(base + 1, v.y);
    atomicAdd(base + 2, v.z);
    atomicAdd(base + 3, v.w);
    if (lane == 0) atomicAdd(cnt + d, 1.0f);
}

// ---- mean  c = sums / max(cnt, 1) --------------------------------------
__global__ __launch_bounds__(256) void mean_kernel(
    const float* __restrict__ sums, const float* __restrict__ cnt,
    float* __restrict__ c, long n) {
    long i = (long)blockIdx.x * blockDim.x + threadIdx.x;
    if (i >= n) return;
    long node = i >> 7;
    c[i] = sums[i] / fmaxf(cnt[node], 1.0f);
}

// ---- GRU cell (R=2): 12 accumulators, K=384 (u=[x|c]) + K=256 (h) ------
__global__ __launch_bounds__(128) void gru_kernel(
    const float* __restrict__ x, const float* __restrict__ c,
    const float* __restrict__ h, const float* __restrict__ pWih,
    const float* __restrict__ pWhh, const float* __restrict__ bih,
    const float* __restrict__ bhh, float* __restrict__ hnew, int N) {
    const int wave = threadIdx.x >> 5;
    const int lane = threadIdx.x & 31;
    const int row0 = (blockIdx.x * 4 + wave) * 32;
    if (row0 >= N) return;
#pragma unroll 1
    for (int nt = 0; nt < 16; ++nt) {
        v8f ir0 = {}, iz0 = {}, in0 = {}, ir1 = {}, iz1 = {}, in1 = {};
        v8f hr0 = {}, hz0 = {}, hn0 = {}, hr1 = {}, hz1 = {}, hn1 = {};
#pragma unroll 2
        for (int kk = 0; kk < 96; ++kk) {            // u @ Wih, K = 384
            int k0 = kk * 4;
            v2f br = load_b_packed(pWih, 96, kk, nt, lane);
            v2f bz = load_b_packed(pWih, 96, kk, nt + 16, lane);
            v2f bn = load_b_packed(pWih, 96, kk, nt + 32, lane);
            v2f a0 = (k0 < 256) ? load_a_frag(x, 256, row0, k0, lane)
                                : load_a_frag(c, 128, row0, k0 - 256, lane);
            v2f a1 = (k0 < 256) ? load_a_frag(x, 256, row0 + 16, k0, lane)
                                : load_a_frag(c, 128, row0 + 16, k0 - 256, lane);
            ir0 = wmma4(a0, br, ir0);  ir1 = wmma4(a1, br, ir1);
            iz0 = wmma4(a0, bz, iz0);  iz1 = wmma4(a1, bz, iz1);
            in0 = wmma4(a0, bn, in0);  in1 = wmma4(a1, bn, in1);
        }
#pragma unroll 2
        for (int kk = 0; kk < 64; ++kk) {            // h @ Whh, K = 256
            int k0 = kk * 4;
            v2f br = load_b_packed(pWhh, 64, kk, nt, lane);
            v2f bz = load_b_packed(pWhh, 64, kk, nt + 16, lane);
            v2f bn = load_b_packed(pWhh, 64, kk, nt + 32, lane);
            v2f a0 = load_a_frag(h, 256, row0, k0, lane);
            v2f a1 = load_a_frag(h, 256, row0 + 16, k0, lane);
            hr0 = wmma4(a0, br, hr0);  hr1 = wmma4(a1, br, hr1);
            hz0 = wmma4(a0, bz, hz0);  hz1 = wmma4(a1, bz, hz1);
            hn0 = wmma4(a0, bn, hn0);  hn1 = wmma4(a1, bn, hn1);
        }
        int col = nt * 16 + (lane & 15);
        float bir = bih[col], biz = bih[col + 256], bin = bih[col + 512];
        float bhr = bhh[col], bhz = bhh[col + 256], bhn = bhh[col + 512];
#pragma unroll
        for (int i = 0; i < 8; ++i) {
            int r = row0 + i + 8 * (lane >> 4);
            {
                float rg = sigmoid_f((ir0[i] + bir) + (hr0[i] + bhr));
                float zg = sigmoid_f((iz0[i] + biz) + (hz0[i] + bhz));
                float ng = tanh_f((in0[i] + bin) + rg * (hn0[i] + bhn));
                float hv = h[(size_t)r * 256 + col];
                hnew[(size_t)r * 256 + col] = (1.0f - zg) * ng + zg * hv;
            }
            {
                float rg = sigmoid_f((ir1[i] + bir) + (hr1[i] + bhr));
                float zg = sigmoid_f((iz1[i] + biz) + (hz1[i] + bhz));
                float ng = tanh_f((in1[i] + bin) + rg * (hn1[i] + bhn));
                float hv = h[(size_t)(r + 16) * 256 + col];
                hnew[(size_t)(r + 16) * 256 + col] = (1.0f - zg) * ng + zg * hv;
            }
        }
    }
}

// ---- head  q = h_new @ out_W + out_b  (R=2, single col tile) -----------
__global__ __launch_bounds__(128) void out_kernel(
    const float* __restrict__ hnew, const float* __restrict__ poutW,
    const float* __restrict__ outb, float* __restrict__ q, int N) {
    const int wave = threadIdx.x >> 5;
    const int lane = threadIdx.x & 31;
    const int row0 = (blockIdx.x * 4 + wave) * 32;
    if (row0 >= N) return;
    v8f acc0 = {}, acc1 = {};
#pragma unroll 4
    for (int kk = 0; kk < 64; ++kk) {
        v2f b = load_b_packed(poutW, 64, kk, 0, lane);
        v2f a0 = load_a_frag(hnew, 256, row0, kk * 4, lane);
        v2f a1 = load_a_frag(hnew, 256, row0 + 16, kk * 4, lane);
        acc0 = wmma4(a0, b, acc0);
        acc1 = wmma4(a1, b, acc1);
    }
    int col = lane & 15;
    float bias = outb[col];
#pragma unroll
    for (int i = 0; i < 8; ++i) {
        int r = row0 + i + 8 * (lane >> 4);
        q[(size_t)r * 16 + col] = acc0[i] + bias;
        q[(size_t)(r + 16) * 16 + col] = acc1[i] + bias;
    }
}

static inline int packBlocks(int K, int Nc) {
    int waves = (K / 4) * (Nc / 16);
    return (waves + 7) / 8;          // 8 waves per 256-thread block
}

extern "C" void kernel_launch(void* const* d_in, const int* in_sizes, int n_in,
                              void* d_out, int out_size, void* d_ws,
                              size_t ws_size, hipStream_t stream) {
    (void)n_in; (void)out_size; (void)ws_size;
    const float* feat = (const float*)d_in[0];
    const float* h    = (const float*)d_in[1];
    const int*   src  = (const int*)d_in[2];
    const int*   dst  = (const int*)d_in[3];
    const float* W0   = (const float*)d_in[4];
    const float* b0   = (const float*)d_in[5];
    const float* W1   = (const float*)d_in[6];
    const float* b1   = (const float*)d_in[7];
    const float* msgW = (const float*)d_in[8];
    const float* msgb = (const float*)d_in[9];
    const float* Wih  = (const float*)d_in[10];
    const float* Whh  = (const float*)d_in[11];
    const float* bih  = (const float*)d_in[12];
    const float* bhh  = (const float*)d_in[13];
    const float* outW = (const float*)d_in[14];
    const float* outb = (const float*)d_in[15];

    const int N = in_sizes[1] / 256;   // 20000
    const int E = in_sizes[2];         // 640000

    float* ws    = (float*)d_ws;
    float* x     = ws;                              // N*256
    float* mx    = x    + (size_t)N * 256;          // N*128
    float* sums  = mx   + (size_t)N * 128;          // N*128
    float* cnt   = sums + (size_t)N * 128;          // N (contiguous w/ sums)
    float* c     = cnt  + N;                        // N*128
    float* pW0   = c    + (size_t)N * 128;          // 16*16*64   = 16384
    float* pW1   = pW0  + 16384;                    // 64*16*64   = 65536
    float* pmsgW = pW1  + 65536;                    // 128*8*64   = 65536
    float* pWih  = pmsgW + 65536;                   // 96*48*64   = 294912
    float* pWhh  = pWih + 294912;                   // 64*48*64   = 196608
    float* poutW = pWhh + 196608;                   // 64*1*64    = 4096

    float* q    = (float*)d_out;                    // N*16
    float* hnew = q + (size_t)N * 16;               // N*256

    const int encBlocks  = (N / 16 + 3) / 4;        // 16-row waves
    const int gemmBlocks = (N / 32 + 3) / 4;        // 32-row waves (R=2)
    const long zeroN     = (long)N * 128 + N;
    const int zeroBlocks = (int)((zeroN + 255) / 256);
    const int edgeBlocks = (E + 7) / 8;
    const long meanN     = (long)N * 128;
    const int meanBlocks = (int)((meanN + 255) / 256);

    // weight packing (fragment-ordered B), ~3 MB total, L2-resident
    pack_kernel<<<packBlocks(64, 256), 256, 0, stream>>>(W0, pW0, 64, 256);
    pack_kernel<<<packBlocks(256, 256), 256, 0, stream>>>(W1, pW1, 256, 256);
    pack_kernel<<<packBlocks(512, 128), 256, 0, stream>>>(msgW, pmsgW, 512, 128);
    pack_kernel<<<packBlocks(384, 768), 256, 0, stream>>>(Wih, pWih, 384, 768);
    pack_kernel<<<packBlocks(256, 768), 256, 0, stream>>>(Whh, pWhh, 256, 768);
    pack_kernel<<<packBlocks(256, 16), 256, 0, stream>>>(outW, poutW, 256, 16);

    zero_kernel<<<zeroBlocks, 256, 0, stream>>>(sums, zeroN);
    enc_kernel<<<encBlocks, 128, 0, stream>>>(feat, pW0, b0, pW1, b1, x, N);
    msg_kernel<<<gemmBlocks, 128, 0, stream>>>(x, h, pmsgW, msgb, mx, N);
    edge_kernel<<<edgeBlocks, 256, 0, stream>>>(src, dst, mx, sums, cnt, E);
    mean_kernel<<<meanBlocks, 256, 0, stream>>>(sums, cnt, c, meanN);
    gru_kernel<<<gemmBlocks, 128, 0, stream>>>(x, c, h, pWih, pWhh, bih, bhh,
                                               hnew, N);
    out_kernel<<<gemmBlocks, 128, 0, stream>>>(hnew, poutW, outb, q, N);
}